// CotSR_25958782337829
// MI455X (gfx1250) — compile-verified
//
#include <hip/hip_runtime.h>

// Problem sizes (fixed by the reference)
#define BB   4
#define CC   96
#define NN   4096      // H*W = 64*64
#define C8P  16        // head dim 12, padded to 16 rows in workspace
#define NCL  96        // num classes

// workspace layout in floats:
//   Qp : [2][B][16][N]   at offset 0
//   Kp : [2][B][16][N]   at offset QPSZ
//   V  : [2][B][96][N]   at offset 2*QPSZ
#define QPSZ (2 * BB * C8P * NN)

typedef float v2f __attribute__((ext_vector_type(2)));
typedef float v8f __attribute__((ext_vector_type(8)));

// pointer types matching the async-to-LDS builtin signatures
typedef int v2i __attribute__((vector_size(8)));
typedef __attribute__((address_space(1))) int*  gi32_p;   // global b32 src
typedef __attribute__((address_space(3))) int*  li32_p;   // LDS b32 dst
typedef __attribute__((address_space(1))) v2i*  gi64_p;   // global b64 src
typedef __attribute__((address_space(3))) v2i*  li64_p;   // LDS b64 dst

#if __has_builtin(__builtin_amdgcn_global_load_async_to_lds_b32) && \
    __has_builtin(__builtin_amdgcn_global_load_async_to_lds_b64) && \
    __has_builtin(__builtin_amdgcn_s_wait_asynccnt)
#define USE_ASYNC 1
#else
#define USE_ASYNC 0
#endif

// -------------------------------------------------------------------------
// Kernel 1: 1x1-conv QKV projections into workspace.
// grid: (N/256, B, 2 streams), block 256. One thread = one pixel n.
// -------------------------------------------------------------------------
__global__ __launch_bounds__(256)
void cotsr_qkv(const float* __restrict__ x1, const float* __restrict__ x2,
               const float* __restrict__ wq1, const float* __restrict__ bq1,
               const float* __restrict__ wk1, const float* __restrict__ bk1,
               const float* __restrict__ wv1, const float* __restrict__ bv1,
               const float* __restrict__ wq2, const float* __restrict__ bq2,
               const float* __restrict__ wk2, const float* __restrict__ bk2,
               const float* __restrict__ wv2, const float* __restrict__ bv2,
               float* __restrict__ ws)
{
    const int s = blockIdx.z, b = blockIdx.y;
    const int n = blockIdx.x * 256 + threadIdx.x;

    const float* x  = (s == 0 ? x1 : x2) + (size_t)b * CC * NN;
    const float* wq = s == 0 ? wq1 : wq2;  const float* bq = s == 0 ? bq1 : bq2;
    const float* wk = s == 0 ? wk1 : wk2;  const float* bk = s == 0 ? bk1 : bk2;
    const float* wv = s == 0 ? wv1 : wv2;  const float* bv = s == 0 ? bv1 : bv2;

    float* Qp = ws +              (size_t)(s * BB + b) * C8P * NN;
    float* Kp = ws + QPSZ +       (size_t)(s * BB + b) * C8P * NN;
    float* V  = ws + 2 * QPSZ +   (size_t)(s * BB + b) * CC  * NN;

    // Q and K rows (12 each), streaming x once
    float aq[12], ak[12];
#pragma unroll
    for (int j = 0; j < 12; ++j) { aq[j] = bq[j]; ak[j] = bk[j]; }
    for (int c = 0; c < CC; ++c) {
        float xv = x[c * NN + n];
#pragma unroll
        for (int j = 0; j < 12; ++j) {
            aq[j] += wq[j * CC + c] * xv;
            ak[j] += wk[j * CC + c] * xv;
        }
    }
#pragma unroll
    for (int j = 0; j < 12; ++j) { Qp[j * NN + n] = aq[j]; Kp[j * NN + n] = ak[j]; }
#pragma unroll
    for (int j = 12; j < 16; ++j) { Qp[j * NN + n] = 0.f; Kp[j * NN + n] = 0.f; }

    // V rows in 4 groups of 24
    for (int g = 0; g < 4; ++g) {
        float av[24];
#pragma unroll
        for (int j = 0; j < 24; ++j) av[j] = bv[g * 24 + j];
        for (int c = 0; c < CC; ++c) {
            float xv = x[c * NN + n];
#pragma unroll
            for (int j = 0; j < 24; ++j) av[j] += wv[(g * 24 + j) * CC + c] * xv;
        }
#pragma unroll
        for (int j = 0; j < 24; ++j) V[(g * 24 + j) * NN + n] = av[j];
    }
}

// -------------------------------------------------------------------------
// Kernel 2: fused flash cross-attention + residual + classifier conv.
// grid: (N/128, B, 2 streams), block 256 (8 waves; one wave = 16 queries).
// K/V tiles double-buffered in LDS and staged with async global->LDS DMA
// (ASYNCcnt) so global latency hides behind the WMMA pipe.
// -------------------------------------------------------------------------
__global__ __launch_bounds__(256)
void cotsr_attn(const float* __restrict__ x1, const float* __restrict__ x2,
                const float* __restrict__ g1, const float* __restrict__ g2,
                const float* __restrict__ wc1, const float* __restrict__ bc1,
                const float* __restrict__ wc2, const float* __restrict__ bc2,
                const float* __restrict__ ws, float* __restrict__ out)
{
    __shared__ __align__(16) float kls[2][16 * 16];   // K tile, transposed [m][c]
    __shared__ __align__(16) float vls[2][CC * 16];   // V tile [c][m]
    __shared__ __align__(16) float wcs[NCL * CC];     // classifier weights, row-major

    const int s = blockIdx.z, b = blockIdx.y, qg = blockIdx.x;
    const int t = 1 - s;                              // kv stream
    const int tid  = threadIdx.x;
    const int lane = tid & 31, wid = tid >> 5;
    const bool hi  = lane >= 16;
    const int  ln  = lane & 15;

    const float* xr    = (s == 0 ? x1 : x2) + (size_t)b * CC * NN;
    const float  gamma = (s == 0 ? g1 : g2)[0];
    const float* wc    = s == 0 ? wc1 : wc2;
    const float* bc    = s == 0 ? bc1 : bc2;
    const float* Qp = ws +            (size_t)(s * BB + b) * C8P * NN;
    const float* Kp = ws + QPSZ +     (size_t)(t * BB + b) * C8P * NN;
    const float* V  = ws + 2 * QPSZ + (size_t)(t * BB + b) * CC  * NN;
    float* outp = out + ((size_t)s * BB + b) * NCL * NN;

    // stage classifier weights once
    for (int i = tid; i < NCL * CC; i += 256) wcs[i] = wc[i];

    const int n0 = qg * 128 + wid * 16;

    // cooperative tile staging (async DMA to LDS when available)
    auto stage = [&](int kt, int buf) {
        const int m0 = kt * 16;
        const int ck = tid >> 4, mk = tid & 15;
#if USE_ASYNC
        // K tile: transpose during the async copy (per-lane LDS destination)
        __builtin_amdgcn_global_load_async_to_lds_b32(
            (gi32_p)(void*)const_cast<float*>(Kp + ck * NN + m0 + mk),
            (li32_p)(void*)&kls[buf][mk * 16 + ck], 0, 0);
#pragma unroll
        for (int i = 0; i < 3; ++i) {
            int idx = tid + 256 * i;
            int cv = idx >> 3, j2 = (idx & 7) * 2;
            __builtin_amdgcn_global_load_async_to_lds_b64(
                (gi64_p)(void*)const_cast<float*>(V + cv * NN + m0 + j2),
                (li64_p)(void*)&vls[buf][cv * 16 + j2], 0, 0);
        }
#else
        kls[buf][mk * 16 + ck] = Kp[ck * NN + m0 + mk];
#pragma unroll
        for (int i = 0; i < 6; ++i) {
            int idx = tid + 256 * i;
            int cv = idx >> 4, m2 = idx & 15;
            vls[buf][cv * 16 + m2] = V[cv * NN + m0 + m2];
        }
#endif
    };

    // Q tile in WMMA-B layout: 3 K-groups (c = 0..11), v2f each
    v2f qb[3];
#pragma unroll
    for (int g = 0; g < 3; ++g) {
        int r0 = 4 * g + (hi ? 2 : 0);
        qb[g][0] = Qp[(r0    ) * NN + n0 + ln];
        qb[g][1] = Qp[(r0 + 1) * NN + n0 + ln];
    }

    v8f oacc[6];                // O[c=96][n=16] fp32 accumulators
#pragma unroll
    for (int i = 0; i < 6; ++i) oacc[i] = (v8f)0.f;
    float rmax = -3.0e38f, rsum = 0.f;

    stage(0, 0);                // prefetch first tile

    for (int kt = 0; kt < 256; ++kt) {
        const int buf = kt & 1;
        if (kt + 1 < 256) {
            stage(kt + 1, buf ^ 1);        // prefetch next tile into other buffer
#if USE_ASYNC
            __builtin_amdgcn_s_wait_asynccnt(4);   // our 4 ops for THIS tile landed
#endif
        } else {
#if USE_ASYNC
            __builtin_amdgcn_s_wait_asynccnt(0);
#endif
        }
        __syncthreads();        // all waves' staging of this tile visible

        // S[m,n] = sum_c K[m,c] * Q[n,c]   (A = K tile, B = Q tile)
        v8f sacc = (v8f)0.f;
#pragma unroll
        for (int g = 0; g < 3; ++g) {
            v2f ka = *(const v2f*)&kls[buf][ln * 16 + 4 * g + (hi ? 2 : 0)];
            sacc = __builtin_amdgcn_wmma_f32_16x16x4_f32(
                false, ka, false, qb[g], (short)0, sacc, false, false);
        }

        // online softmax over m (in-lane over 8 regs + one cross-half shuffle)
        float tmax = sacc[0];
#pragma unroll
        for (int r = 1; r < 8; ++r) tmax = fmaxf(tmax, sacc[r]);
        tmax = fmaxf(tmax, __shfl_xor(tmax, 16, 32));
        float nmax = fmaxf(rmax, tmax);
        float corr = __expf(rmax - nmax);
        rmax = nmax;
        float p[8], tsum = 0.f;
#pragma unroll
        for (int r = 0; r < 8; ++r) { p[r] = __expf(sacc[r] - nmax); tsum += p[r]; }
        tsum += __shfl_xor(tsum, 16, 32);
        rsum = rsum * corr + tsum;
#pragma unroll
        for (int i = 0; i < 6; ++i)
#pragma unroll
            for (int r = 0; r < 8; ++r) oacc[i][r] *= corr;

        // rearrange P[m,n] into B-layout (4 K-groups over m)
        float xp[8];
#pragma unroll
        for (int r = 0; r < 8; ++r) xp[r] = __shfl_xor(p[r], 16, 32);
        v2f pb[4];
        pb[0][0] = hi ? xp[2] : p[0];  pb[0][1] = hi ? xp[3] : p[1];
        pb[1][0] = hi ? xp[6] : p[4];  pb[1][1] = hi ? xp[7] : p[5];
        pb[2][0] = hi ? p[2]  : xp[0]; pb[2][1] = hi ? p[3]  : xp[1];
        pb[3][0] = hi ? p[6]  : xp[4]; pb[3][1] = hi ? p[7]  : xp[5];

        // O[c,n] += sum_m V[c,m] * P[m,n]
#pragma unroll
        for (int cc = 0; cc < 6; ++cc) {
#pragma unroll
            for (int g = 0; g < 4; ++g) {
                v2f va = *(const v2f*)&vls[buf][(cc * 16 + ln) * 16 + 4 * g + (hi ? 2 : 0)];
                oacc[cc] = __builtin_amdgcn_wmma_f32_16x16x4_f32(
                    false, va, false, pb[g], (short)0, oacc[cc], false, false);
            }
        }
        __syncthreads();        // all waves done reading before buffer reuse
    }

    // epilogue: y = x + gamma * O / rsum
    const float inv = 1.0f / rsum;
#pragma unroll
    for (int cc = 0; cc < 6; ++cc) {
#pragma unroll
        for (int r = 0; r < 8; ++r) {
            int c = cc * 16 + r + (hi ? 8 : 0);
            float xv = xr[c * NN + n0 + ln];
            oacc[cc][r] = xv + gamma * oacc[cc][r] * inv;
        }
    }

    // classifier: out[o,n] = sum_c wc[o,c] * y[c,n] + bc[o]
    v8f res[6];
#pragma unroll
    for (int i = 0; i < 6; ++i) res[i] = (v8f)0.f;

#pragma unroll
    for (int cc = 0; cc < 6; ++cc) {
        float yy[8], xy[8];
#pragma unroll
        for (int r = 0; r < 8; ++r) { yy[r] = oacc[cc][r]; xy[r] = __shfl_xor(yy[r], 16, 32); }
        v2f yb[4];
        yb[0][0] = hi ? xy[2] : yy[0];  yb[0][1] = hi ? xy[3] : yy[1];
        yb[1][0] = hi ? xy[6] : yy[4];  yb[1][1] = hi ? xy[7] : yy[5];
        yb[2][0] = hi ? yy[2] : xy[0];  yb[2][1] = hi ? yy[3] : xy[1];
        yb[3][0] = hi ? yy[6] : xy[4];  yb[3][1] = hi ? yy[7] : xy[5];
#pragma unroll
        for (int oc = 0; oc < 6; ++oc) {
#pragma unroll
            for (int g = 0; g < 4; ++g) {
                v2f wa = *(const v2f*)&wcs[(oc * 16 + ln) * CC + cc * 16 + 4 * g + (hi ? 2 : 0)];
                res[oc] = __builtin_amdgcn_wmma_f32_16x16x4_f32(
                    false, wa, false, yb[g], (short)0, res[oc], false, false);
            }
        }
    }

#pragma unroll
    for (int oc = 0; oc < 6; ++oc) {
#pragma unroll
        for (int r = 0; r < 8; ++r) {
            int o = oc * 16 + r + (hi ? 8 : 0);
            outp[o * NN + n0 + ln] = res[oc][r] + bc[o];
        }
    }
}

// -------------------------------------------------------------------------
extern "C" void kernel_launch(void* const* d_in, const int* in_sizes, int n_in,
                              void* d_out, int out_size, void* d_ws, size_t ws_size,
                              hipStream_t stream)
{
    const float* x1  = (const float*)d_in[0];
    const float* x2  = (const float*)d_in[1];
    const float* wq1 = (const float*)d_in[2];
    const float* bq1 = (const float*)d_in[3];
    const float* wk1 = (const float*)d_in[4];
    const float* bk1 = (const float*)d_in[5];
    const float* wv1 = (const float*)d_in[6];
    const float* bv1 = (const float*)d_in[7];
    const float* wq2 = (const float*)d_in[8];
    const float* bq2 = (const float*)d_in[9];
    const float* wk2 = (const float*)d_in[10];
    const float* bk2 = (const float*)d_in[11];
    const float* wv2 = (const float*)d_in[12];
    const float* bv2 = (const float*)d_in[13];
    const float* g1  = (const float*)d_in[14];
    const float* g2  = (const float*)d_in[15];
    const float* wc1 = (const float*)d_in[16];
    const float* bc1 = (const float*)d_in[17];
    const float* wc2 = (const float*)d_in[18];
    const float* bc2 = (const float*)d_in[19];

    float* ws  = (float*)d_ws;
    float* out = (float*)d_out;

    dim3 gq(NN / 256, BB, 2);
    cotsr_qkv<<<gq, 256, 0, stream>>>(x1, x2, wq1, bq1, wk1, bk1, wv1, bv1,
                                      wq2, bq2, wk2, bk2, wv2, bv2, ws);

    dim3 ga(NN / 128, BB, 2);
    cotsr_attn<<<ga, 256, 0, stream>>>(x1, x2, g1, g2, wc1, bc1, wc2, bc2, ws, out);
}